// GNN_31937376813124
// MI455X (gfx1250) — compile-verified
//
#include <hip/hip_runtime.h>
#include <math.h>

// ---------------------------------------------------------------------------
// Problem constants (match reference)
// ---------------------------------------------------------------------------
#define NN   500000
#define EE   2000000
#define GG   64
#define IND  32
#define EMB  64
#define EPSV 1e-5f

typedef __attribute__((ext_vector_type(2))) float v2f;
typedef __attribute__((ext_vector_type(8))) float v8f;

// ---------------------------------------------------------------------------
// small helpers
// ---------------------------------------------------------------------------
__device__ __forceinline__ void atomicMaxF(float* addr, float v) {
    if (v >= 0.0f) atomicMax((int*)addr, __float_as_int(v));
    else           atomicMin((unsigned int*)addr, __float_as_uint(v));
}

// ---------------------------------------------------------------------------
// 0) init degree to 1.0 (self loops)
// ---------------------------------------------------------------------------
__global__ void k_init_deg(float* __restrict__ deg, int n) {
    int i = blockIdx.x * blockDim.x + threadIdx.x;
    if (i < n) deg[i] = 1.0f;
}

// stats region layout: [0:64]=sum1 [64:128]=sumsq1 [128:192]=sum2 [192:256]=sumsq2
//                      [256:4352]=gmax [4352:8448]=gsum [8448:8512]=gcnt
#define STATS_LEN 8512
__global__ void k_init_stats(float* __restrict__ stats) {
    int i = blockIdx.x * blockDim.x + threadIdx.x;
    if (i >= STATS_LEN) return;
    bool isMax = (i >= 256) && (i < 256 + GG * EMB);
    stats[i] = isMax ? -INFINITY : 0.0f;
}

// ---------------------------------------------------------------------------
// 1) degree accumulate over E edge destinations
// ---------------------------------------------------------------------------
__global__ void k_deg_accum(const int* __restrict__ dst, float* __restrict__ deg, int e) {
    int i = blockIdx.x * blockDim.x + threadIdx.x;
    if (i < e) atomicAdd(&deg[dst[i]], 1.0f);
}

__global__ void k_rsqrt_inplace(float* __restrict__ deg, int n) {
    int i = blockIdx.x * blockDim.x + threadIdx.x;
    if (i < n) deg[i] = rsqrtf(deg[i]);
}

// ---------------------------------------------------------------------------
// 2) node encoder: h[i, 0:16] = emb_type[node_type[i]], h[i, 16:32] = emb_inv[...]
// ---------------------------------------------------------------------------
__global__ void k_encoder(const int* __restrict__ node_type,
                          const int* __restrict__ num_inv,
                          const float* __restrict__ emb_type,
                          const float* __restrict__ emb_inv,
                          float* __restrict__ h, int n) {
    int idx = blockIdx.x * blockDim.x + threadIdx.x;
    if (idx >= n * IND) return;
    int i = idx >> 5;          // node
    int j = idx & 31;          // feature
    float v;
    if (j < 16) v = emb_type[node_type[i] * 16 + j];
    else        v = emb_inv [num_inv [i] * 16 + (j - 16)];
    h[(size_t)i * IND + j] = v;
}

// ---------------------------------------------------------------------------
// 3) WMMA fp32 GEMM:  Out[N,64] = A[N,K] @ W[K,64]
//    wave32: one wave per 16-row tile; 4 col-tiles of 16; v_wmma_f32_16x16x4_f32
//    VGPR layouts per CDNA5 ISA 7.12.2:
//      A 16x4 f32 : lane(0..15)=row M, vgpr{0,1} = K{0,1}; lanes 16..31 = K{2,3}
//      B 4x16 f32 : vgpr r, lanes0-15 -> K=r,   lanes16-31 -> K=r+2, N = lane%16
//      D 16x16 f32: vgpr r, lanes0-15 -> M=r,   lanes16-31 -> M=r+8, N = lane%16
//    W is pre-swizzled into LDS in B-fragment order so each lane's (b.x,b.y)
//    pair is one contiguous, 8B-aligned ds_load_b64 (no repacking movs).
//    LDS index: [k0/4][ntile][lane] -> float2{ W[k][c], W[k+1][c] }
//      with k = k0 + 2*(lane>>4), c = ntile*16 + (lane&15)
// ---------------------------------------------------------------------------
template <int K>
__global__ void k_gemm_wmma(const float* __restrict__ A,
                            const float* __restrict__ W,
                            float* __restrict__ Out, int nrows) {
    __shared__ float2 Ws[(K / 4) * 4 * 32];   // == K*32 float2 == K*64 floats
    for (int idx = threadIdx.x; idx < K * 32; idx += blockDim.x) {
        int kk    = idx >> 7;        // k-step (k0/4)
        int rem   = idx & 127;
        int n     = rem >> 5;        // col tile
        int lane2 = rem & 31;
        int lh2   = lane2 & 15;
        int hi2   = lane2 >> 4;
        int k     = kk * 4 + 2 * hi2;
        int c     = n * 16 + lh2;
        Ws[idx] = make_float2(W[k * EMB + c], W[(k + 1) * EMB + c]);
    }
    __syncthreads();

    const int wave = threadIdx.x >> 5;
    const int lane = threadIdx.x & 31;
    const int tile = blockIdx.x * (blockDim.x >> 5) + wave;
    const int row_base = tile * 16;
    if (row_base >= nrows) return;

    const int lh = lane & 15;   // row within tile (A/D) / col within tile (B/D)
    const int hi = lane >> 4;   // lane-group select

    v8f acc[4] = {v8f{}, v8f{}, v8f{}, v8f{}};
    const float* Arow = A + (size_t)(row_base + lh) * K;

#pragma unroll
    for (int k0 = 0; k0 < K; k0 += 4) {
        v2f a;
        a.x = Arow[k0 + 2 * hi + 0];
        a.y = Arow[k0 + 2 * hi + 1];
#pragma unroll
        for (int n = 0; n < 4; ++n) {
            float2 bb = Ws[(k0 >> 2) * 128 + n * 32 + lane];
            v2f b;
            b.x = bb.x;
            b.y = bb.y;
            acc[n] = __builtin_amdgcn_wmma_f32_16x16x4_f32(
                false, a, false, b, (short)0, acc[n], false, false);
        }
    }

#pragma unroll
    for (int n = 0; n < 4; ++n) {
#pragma unroll
        for (int r = 0; r < 8; ++r) {
            Out[(size_t)(row_base + r + hi * 8) * EMB + n * 16 + lh] = acc[n][r];
        }
    }
}

// ---------------------------------------------------------------------------
// 4) edge scatter: acc[dst] += hw[src] * dis[src]*dis[dst]
//    16 lanes per edge, float4 per lane, f32 atomics (stay in 192MB L2)
// ---------------------------------------------------------------------------
__global__ void k_scatter(const int* __restrict__ src, const int* __restrict__ dst,
                          const float* __restrict__ dis,
                          const float* __restrict__ hw, float* __restrict__ acc, int e) {
    long long idx = (long long)blockIdx.x * blockDim.x + threadIdx.x;
    int eid = (int)(idx >> 4);
    if (eid >= e) return;
    int f4 = ((int)idx & 15) * 4;
    int s = src[eid], d = dst[eid];
    float norm = dis[s] * dis[d];
    const float4 v = *reinterpret_cast<const float4*>(hw + (size_t)s * EMB + f4);
    float* p = acc + (size_t)d * EMB + f4;
    atomicAdd(p + 0, v.x * norm);
    atomicAdd(p + 1, v.y * norm);
    atomicAdd(p + 2, v.z * norm);
    atomicAdd(p + 3, v.w * norm);
}

// ---------------------------------------------------------------------------
// 5) fused self-loop + bias + BN statistics (sum / sumsq per channel)
// ---------------------------------------------------------------------------
#define RPB 512   // rows per block
__global__ void k_finish_stats(float* __restrict__ acc, const float* __restrict__ hw,
                               const float* __restrict__ dis, const float* __restrict__ bias,
                               float* __restrict__ sum, float* __restrict__ sumsq, int n) {
    const int j = threadIdx.x & 63;     // channel
    const int rs = threadIdx.x >> 6;    // 0..3 row sub-group
    const int row0 = blockIdx.x * RPB;
    const float bj = bias[j];
    float s = 0.0f, ss = 0.0f;
    int rend = row0 + RPB; if (rend > n) rend = n;
    for (int r = row0 + rs; r < rend; r += 4) {
        float d = dis[r];
        float selfn = d * d;            // 1/deg
        size_t off = (size_t)r * EMB + j;
        float v = acc[off] + hw[off] * selfn + bj;
        acc[off] = v;
        s += v; ss += v * v;
    }
    __shared__ float ls[256];
    __shared__ float lss[256];
    ls[threadIdx.x] = s; lss[threadIdx.x] = ss;
    __syncthreads();
    if (rs == 0) {
        s  = ls [j] + ls [64 + j] + ls [128 + j] + ls [192 + j];
        ss = lss[j] + lss[64 + j] + lss[128 + j] + lss[192 + j];
        atomicAdd(&sum[j], s);
        atomicAdd(&sumsq[j], ss);
    }
}

// ---------------------------------------------------------------------------
// 6) BN apply (+optional relu, +optional fused pooling)
// ---------------------------------------------------------------------------
__global__ void k_bn_apply(float* __restrict__ x,
                           const float* __restrict__ sum, const float* __restrict__ sumsq,
                           const float* __restrict__ gamma, const float* __restrict__ beta,
                           int n, int do_relu, int do_pool,
                           const int* __restrict__ batch,
                           float* __restrict__ gmax, float* __restrict__ gsum,
                           float* __restrict__ gcnt) {
    long long idx = (long long)blockIdx.x * blockDim.x + threadIdx.x;
    if (idx >= (long long)n * EMB) return;
    int i = (int)(idx >> 6);
    int j = (int)idx & 63;
    float mu  = sum[j] * (1.0f / (float)NN);
    float var = sumsq[j] * (1.0f / (float)NN) - mu * mu;
    float v = (x[idx] - mu) * rsqrtf(var + EPSV) * gamma[j] + beta[j];
    if (do_relu) v = fmaxf(v, 0.0f);
    x[idx] = v;
    if (do_pool) {
        int g = batch[i];
        atomicMaxF(&gmax[g * EMB + j], v);
        atomicAdd(&gsum[g * EMB + j], v);
        if (j == 0) atomicAdd(&gcnt[g], 1.0f);
    }
}

// ---------------------------------------------------------------------------
// 7) final output: [G, 2*EMB] = concat(max, mean)
// ---------------------------------------------------------------------------
__global__ void k_final(const float* __restrict__ gmax, const float* __restrict__ gsum,
                        const float* __restrict__ gcnt, float* __restrict__ out) {
    int idx = blockIdx.x * blockDim.x + threadIdx.x;
    if (idx >= GG * EMB) return;
    int g = idx >> 6;
    int j = idx & 63;
    out[g * (2 * EMB) + j]       = gmax[g * EMB + j];
    out[g * (2 * EMB) + EMB + j] = gsum[g * EMB + j] / fmaxf(gcnt[g], 1.0f);
}

// ---------------------------------------------------------------------------
// launch
// ---------------------------------------------------------------------------
extern "C" void kernel_launch(void* const* d_in, const int* in_sizes, int n_in,
                              void* d_out, int out_size, void* d_ws, size_t ws_size,
                              hipStream_t stream) {
    const int*   node_type = (const int*)d_in[0];
    const int*   num_inv   = (const int*)d_in[1];
    const int*   e_src     = (const int*)d_in[2];         // edge_index row 0
    const int*   e_dst     = e_src + EE;                  // edge_index row 1
    const int*   batch     = (const int*)d_in[3];
    const float* emb_type  = (const float*)d_in[4];
    const float* emb_inv   = (const float*)d_in[5];
    const float* W1        = (const float*)d_in[6];
    const float* b1        = (const float*)d_in[7];
    const float* W2        = (const float*)d_in[8];
    const float* b2        = (const float*)d_in[9];
    const float* gamma1    = (const float*)d_in[10];
    const float* beta1     = (const float*)d_in[11];
    const float* gamma2    = (const float*)d_in[12];
    const float* beta2     = (const float*)d_in[13];
    float* out = (float*)d_out;

    // workspace layout (floats)
    float* ws    = (float*)d_ws;
    float* dis   = ws;                                 // N   (deg -> rsqrt(deg))
    float* h     = dis + (size_t)NN;                   // N*32
    float* hw    = h   + (size_t)NN * IND;             // N*64 (GEMM out / self-loop src)
    float* acc   = hw  + (size_t)NN * EMB;             // N*64 (scatter accum / BN in place)
    float* stats = acc + (size_t)NN * EMB;             // STATS_LEN
    float* sum1 = stats, *sumsq1 = stats + 64, *sum2 = stats + 128, *sumsq2 = stats + 192;
    float* gmax = stats + 256;
    float* gsum = gmax + GG * EMB;
    float* gcnt = gsum + GG * EMB;

    const int T = 256;
    // --- degree / norm ---
    k_init_deg<<<(NN + T - 1) / T, T, 0, stream>>>(dis, NN);
    k_init_stats<<<(STATS_LEN + T - 1) / T, T, 0, stream>>>(stats);
    k_deg_accum<<<(EE + T - 1) / T, T, 0, stream>>>(e_dst, dis, EE);
    k_rsqrt_inplace<<<(NN + T - 1) / T, T, 0, stream>>>(dis, NN);

    // --- encoder ---
    k_encoder<<<(NN * IND + T - 1) / T, T, 0, stream>>>(node_type, num_inv, emb_type, emb_inv, h, NN);

    const int tiles  = (NN + 15) / 16;
    const int gblk   = (tiles + 7) / 8;          // 8 waves / 256-thread block
    const long long sthreads = (long long)EE * 16;
    const int sblk = (int)((sthreads + T - 1) / T);
    const long long bthreads = (long long)NN * EMB;
    const int bblk = (int)((bthreads + T - 1) / T);

    // --- layer 1 ---
    k_gemm_wmma<IND><<<gblk, T, 0, stream>>>(h, W1, hw, NN);
    hipMemsetAsync(acc, 0, (size_t)NN * EMB * sizeof(float), stream);
    k_scatter<<<sblk, T, 0, stream>>>(e_src, e_dst, dis, hw, acc, EE);
    k_finish_stats<<<(NN + RPB - 1) / RPB, T, 0, stream>>>(acc, hw, dis, b1, sum1, sumsq1, NN);
    k_bn_apply<<<bblk, T, 0, stream>>>(acc, sum1, sumsq1, gamma1, beta1, NN,
                                       /*relu=*/1, /*pool=*/0, nullptr, nullptr, nullptr, nullptr);

    // --- layer 2 ---
    k_gemm_wmma<EMB><<<gblk, T, 0, stream>>>(acc, W2, hw, NN);
    hipMemsetAsync(acc, 0, (size_t)NN * EMB * sizeof(float), stream);
    k_scatter<<<sblk, T, 0, stream>>>(e_src, e_dst, dis, hw, acc, EE);
    k_finish_stats<<<(NN + RPB - 1) / RPB, T, 0, stream>>>(acc, hw, dis, b2, sum2, sumsq2, NN);
    k_bn_apply<<<bblk, T, 0, stream>>>(acc, sum2, sumsq2, gamma2, beta2, NN,
                                       /*relu=*/0, /*pool=*/1, batch, gmax, gsum, gcnt);

    // --- output ---
    k_final<<<(GG * EMB + T - 1) / T, T, 0, stream>>>(gmax, gsum, gcnt, out);
}